// GraphBuilder_61254823575573
// MI455X (gfx1250) — compile-verified
//
#include <hip/hip_runtime.h>
#include <math.h>

#define NPTS 4096
#define BATCH 4
#define KNN 16
#define LOGN 12  // NPTS = 1<<12

typedef __attribute__((ext_vector_type(2))) float v2f;
typedef __attribute__((ext_vector_type(8))) float v8f;

// ---------------- zero-fill the dense Laplacian (268 MB streaming write) ----
__global__ void __launch_bounds__(256) zero_kernel(float4* __restrict__ out, long long n4) {
    long long i = (long long)blockIdx.x * 256 + threadIdx.x;
    if (i < n4) out[i] = make_float4(0.f, 0.f, 0.f, 0.f);
}

// ---------------- diag = 1, degree accumulator = 0 --------------------------
__global__ void __launch_bounds__(256) init_small_kernel(float* __restrict__ L,
                                                         float* __restrict__ Dacc) {
    int i = blockIdx.x * 256 + threadIdx.x;
    if (i < BATCH * NPTS) {
        Dacc[i] = 0.f;
        int b = i >> LOGN;
        int r = i & (NPTS - 1);
        L[((size_t)b * NPTS + r) * NPTS + r] = 1.0f;
    }
}

// ---------------- WMMA-based KNN + Gaussian weights + degree atomics --------
// block = 256 threads (8 waves), handles 128 rows of one batch.
// wave handles 16 rows; one V_WMMA_F32_16X16X4_F32 per 16x16 distance tile:
//   A[m] = (-2x, -2y, -2z, |p_m|^2),  B[:,n] = (x', y', z', 1),  C[m,n] = |p_n|^2
//   => D[m,n] = |p_m|^2 + |p_n|^2 - 2 p_m.p_n  (exact d^2 tile in one op)
__global__ void __launch_bounds__(256) knn_kernel(const float* __restrict__ xyz,
                                                  int*   __restrict__ iOut,
                                                  float* __restrict__ wOut,
                                                  float* __restrict__ Dacc) {
    __shared__ float sX[NPTS], sY[NPTS], sZ[NPTS];   // 48 KB candidate coords
    __shared__ float sTile[8][256];                  // 1 KB per wave d^2 tile

    const int b   = blockIdx.y;
    const int tid = threadIdx.x;
    const float* xb = xyz + (size_t)b * 3 * NPTS;    // layout (B,3,N)
    for (int i = tid; i < NPTS; i += 256) {
        sX[i] = xb[i];
        sY[i] = xb[NPTS + i];
        sZ[i] = xb[2 * NPTS + i];
    }
    __syncthreads();

    const int lane = tid & 31;
    const int wave = tid >> 5;
    const int h    = lane >> 4;     // K-half:  k = vgpr + 2*h
    const int mr   = lane & 15;     // N (store phase) / row (select phase)
    const int rowStart = blockIdx.x * 128 + wave * 16;

    // A operand (constant across the tile loop)
    const int rp = rowStart + mr;
    const float px = sX[rp], py = sY[rp], pz = sZ[rp];
    const float psq = px * px + py * py + pz * pz;
    v2f aA;
    aA.x = h ? (-2.0f * pz) : (-2.0f * px);   // k = 2h
    aA.y = h ? psq          : (-2.0f * py);   // k = 2h+1

    float bestd[16];
    int   besti[16];
#pragma unroll
    for (int j = 0; j < 16; ++j) { bestd[j] = 3.0e38f; besti[j] = 0; }

    for (int t = 0; t < NPTS / 16; ++t) {
        const int col = t * 16 + mr;
        const float qx = sX[col], qy = sY[col], qz = sZ[col];
        const float qsq = qx * qx + qy * qy + qz * qz;
        v2f bB;
        bB.x = h ? qz   : qx;                 // k = 2h
        bB.y = h ? 1.0f : qy;                 // k = 2h+1
        v8f c = {qsq, qsq, qsq, qsq, qsq, qsq, qsq, qsq};  // C[m,n] = |p_n|^2

        v8f d = __builtin_amdgcn_wmma_f32_16x16x4_f32(
            false, aA, false, bB, (short)0, c, false, false);

        // D layout: vgpr v, lane l -> (M = v + 8*(l/16), N = l%16)
#pragma unroll
        for (int v = 0; v < 8; ++v)
            sTile[wave][mr * 16 + (v + 8 * h)] = d[v];   // [n][m], 2x ds_store_b128
        __asm__ volatile("s_wait_dscnt 0" ::: "memory");

        // selection: lane owns row mr, columns [8h, 8h+8) of this tile
#pragma unroll
        for (int i = 0; i < 8; ++i) {
            const int cc = 8 * h + i;
            const float dv = sTile[wave][cc * 16 + mr];
            if (dv < bestd[15]) {            // cheap guard; insert itself is branchless
                float cd = dv;
                int   ci = t * 16 + cc;
#pragma unroll
                for (int j = 0; j < 16; ++j) {
                    // branchless conditional swap: keeps list sorted ascending,
                    // displaced max bubbles onward (v_cndmask, no exec branches)
                    const bool  sw = cd < bestd[j];
                    const float nd = sw ? cd       : bestd[j];
                    const float xd = sw ? bestd[j] : cd;
                    const int   ni = sw ? ci       : besti[j];
                    const int   xi = sw ? besti[j] : ci;
                    bestd[j] = nd; cd = xd;
                    besti[j] = ni; ci = xi;
                }
            }
        }
    }

    // merge the two sorted half-lists (lanes l and l^16 hold disjoint column
    // halves of the same row): low half of bitonic merge = min(a[i], b[15-i])
    float md[16]; int mi[16];
#pragma unroll
    for (int i = 0; i < 16; ++i) {
        float pd = __shfl_xor(bestd[15 - i], 16, 32);
        int   pj = __shfl_xor(besti[15 - i], 16, 32);
        bool take = pd < bestd[i];
        md[i] = take ? pd : bestd[i];
        mi[i] = take ? pj : besti[i];
    }

    if (lane < 16) {                          // one lane per row finalizes
        const int r = rowStart + lane;
        const float rx = sX[r], ry = sY[r], rz = sZ[r];
        const size_t base = ((size_t)b * NPTS + r) * KNN;
        float own = 0.f;
#pragma unroll
        for (int i = 0; i < 16; ++i) {
            const int j = mi[i];
            const float dx = rx - sX[j], dy = ry - sY[j], dz = rz - sZ[j];
            const float dist2 = dx * dx + dy * dy + dz * dz;  // exact recompute (as reference)
            const float w = expf(-0.5f * dist2);
            iOut[base + i] = j;
            wOut[base + i] = w;
            own += 0.5f * w;                                  // A[r,j] half-edge
            atomicAdd(&Dacc[b * NPTS + j], 0.5f * w);         // A[j,r] half-edge
        }
        atomicAdd(&Dacc[b * NPTS + r], own);
    }
}

// ---------------- d_is = (max(D, eps))^-1/2 ---------------------------------
__global__ void __launch_bounds__(256) dis_kernel(const float* __restrict__ Dacc,
                                                  float* __restrict__ dis) {
    int i = blockIdx.x * 256 + threadIdx.x;
    if (i < BATCH * NPTS) {
        float Dv = fmaxf(Dacc[i], 1e-6f);
        dis[i] = 1.0f / sqrtf(Dv);
    }
}

// ---------------- symmetric sparse scatter: L -= d_i * 0.5w * d_j -----------
__global__ void __launch_bounds__(256) scatter_kernel(const int*   __restrict__ iIdx,
                                                      const float* __restrict__ wIn,
                                                      const float* __restrict__ dis,
                                                      float* __restrict__ L) {
    int e = blockIdx.x * 256 + threadIdx.x;
    if (e >= BATCH * NPTS * KNN) return;
    int pi = e >> 4;                 // b*N + i
    int bN = pi & ~(NPTS - 1);       // b*N
    int i  = pi & (NPTS - 1);
    int j  = iIdx[e];
    float v = 0.5f * wIn[e] * dis[pi] * dis[bN + j];
    atomicAdd(&L[(size_t)pi * NPTS + j], -v);                 // L[b,i,j]
    atomicAdd(&L[((size_t)bN + j) * NPTS + i], -v);           // L[b,j,i]
}

extern "C" void kernel_launch(void* const* d_in, const int* in_sizes, int n_in,
                              void* d_out, int out_size, void* d_ws, size_t ws_size,
                              hipStream_t stream) {
    const float* xyz = (const float*)d_in[0];
    float* L = (float*)d_out;

    char* ws = (char*)d_ws;
    const size_t edges = (size_t)BATCH * NPTS * KNN;          // 262144
    int*   iIdx = (int*)ws;                                   // 1 MB
    float* wArr = (float*)(ws + edges * sizeof(int));         // 1 MB
    float* Dacc = (float*)(ws + 2 * edges * sizeof(int));     // 64 KB
    float* dis  = Dacc + BATCH * NPTS;                        // 64 KB

    long long n4 = (long long)out_size / 4;                   // float4 count
    zero_kernel<<<(unsigned)((n4 + 255) / 256), 256, 0, stream>>>((float4*)d_out, n4);
    init_small_kernel<<<(BATCH * NPTS + 255) / 256, 256, 0, stream>>>(L, Dacc);

    dim3 g(NPTS / 128, BATCH);
    knn_kernel<<<g, 256, 0, stream>>>(xyz, iIdx, wArr, Dacc);

    dis_kernel<<<(BATCH * NPTS + 255) / 256, 256, 0, stream>>>(Dacc, dis);
    scatter_kernel<<<(unsigned)((edges + 255) / 256), 256, 0, stream>>>(iIdx, wArr, dis, L);
}